// REAFTLoss_82952998355421
// MI455X (gfx1250) — compile-verified
//
#include <hip/hip_runtime.h>
#include <stdint.h>

// REAFT loss, MI455X (gfx1250, wave32).
// B=2, S=2048, V=32000. Rows after shift: 2*2047 = 4094, each 32000 f32 = 128 KB.
// Strategy: one workgroup per row. The CDNA5 Tensor Data Mover copies the whole
// row global->LDS with a single tensor_load_to_lds (TENSORcnt-tracked DMA),
// then two float4 passes out of LDS (max; then sum-exp + entropy sum).
// HBM traffic = 1x (524 MB ~= 22.5 us at 23.3 TB/s). Deterministic reduction via
// per-row partials in d_ws + a second tiny reduction kernel (no float atomics).

#define V_DIM   32000
#define SEQ     2048
#define SROW    2047            // rows per batch after shift
#define BATCH   2
#define NROWS   (BATCH * SROW)  // 4094
#define TPB     256
#define NV4     (V_DIM / 4)     // 8000 float4 chunks per row

#define SHOCK_THRESHOLD 5.0f
#define ALPHA           3.0f
#define EPS_W           1e-6f
#define LOG_V           10.373491181782746f   // ln(32000)

typedef uint32_t u32x4 __attribute__((ext_vector_type(4)));
typedef uint32_t u32x8 __attribute__((ext_vector_type(8)));

__device__ __forceinline__ float wave_max32(float v) {
#pragma unroll
  for (int o = 16; o > 0; o >>= 1) v = fmaxf(v, __shfl_xor(v, o, 32));
  return v;
}
__device__ __forceinline__ float wave_sum32(float v) {
#pragma unroll
  for (int o = 16; o > 0; o >>= 1) v += __shfl_xor(v, o, 32);
  return v;
}

__global__ __launch_bounds__(TPB)
void reaft_row_kernel(const float* __restrict__ logits,
                      const int*   __restrict__ labels,
                      const int*   __restrict__ amask,
                      float*       __restrict__ ws_sum,
                      float*       __restrict__ ws_cnt)
{
  __shared__ __align__(16) float buf[V_DIM];   // 128000 B: whole row staged in LDS
  __shared__ float redA[8];
  __shared__ float redB[8];
  __shared__ float redC[8];

  const int row  = blockIdx.x;
  const int b    = row / SROW;
  const int s    = row % SROW;          // shift_logits row = logits[b, s, :]
  const int tid  = threadIdx.x;
  const int lane = tid & 31;
  const int wid  = tid >> 5;

  const float* src = logits + ((size_t)b * SEQ + (size_t)s) * V_DIM;

  // ---- CDNA5 TDM: one tensor_load_to_lds moves the whole 128 KB row.  ----
  // Tensor ops ignore EXEC (issue even when EXEC==0), so gate the issue
  // behind a *scalar* branch: only wave 0's instruction stream contains it.
  const int wave0 = (__builtin_amdgcn_readfirstlane(tid) >> 5);
  if (wave0 == 0) {
    const uint64_t gaddr = (uint64_t)(uintptr_t)src;
    const uint32_t lds0  = (uint32_t)(uintptr_t)(void*)buf;  // LDS byte offset

    // D# group 0 (ISA 08 §8.3): count=1 | lds_addr | global_addr[56:0] | type=2
    u32x4 g0;
    g0.x = 1u;                                        // count=1, user desc
    g0.y = lds0;                                      // lds_addr (bytes)
    g0.z = (uint32_t)gaddr;                           // global_addr[31:0]
    g0.w = (uint32_t)((gaddr >> 32) & 0x01FFFFFFull)  // global_addr[56:32]
           | (2u << 30);                              // type=2 ("image")

    // D# group 1 (§8.4): 2-D tensor, 1 row x 32000 elements of 4 bytes.
    u32x8 g1;
    g1[0] = 2u << 16;                                 // data_size=2 (4 B), mask=0
    g1[1] = ((uint32_t)V_DIM & 0xFFFFu) << 16;        // tensor_dim0[15:0]
    g1[2] = 1u << 16;                                 // tensor_dim0[31:16]=0, tensor_dim1=1
    g1[3] = ((uint32_t)V_DIM & 0xFFFFu) << 16;        // tile_dim0 = 32000
    g1[4] = 1u;                                       // tile_dim1=1, tile_dim2=0
    g1[5] = (uint32_t)V_DIM;                          // tensor_dim0_stride[31:0]
    g1[6] = ((uint32_t)V_DIM & 0xFFFFu) << 16;        // dim0_stride[47:32]=0, dim1_stride[15:0]
    g1[7] = 0u;                                       // dim1_stride[47:16]=0

    asm volatile("tensor_load_to_lds %0, %1, null, null"
                 :: "s"(g0), "s"(g1) : "memory");
    __builtin_amdgcn_s_wait_tensorcnt(0);
  }
  __syncthreads();   // publish LDS row to all 8 waves

  const float4* b4 = (const float4*)buf;

  // ---- Pass 1: row max (ds_load_b128, 16 B/lane stride -> conflict-free) ----
  float m = -3.402823466e38f;
  for (int i = tid; i < NV4; i += TPB) {
    float4 v = b4[i];
    m = fmaxf(m, fmaxf(fmaxf(v.x, v.y), fmaxf(v.z, v.w)));
  }
  m = wave_max32(m);
  if (lane == 0) redA[wid] = m;
  __syncthreads();
  m = redA[0];
#pragma unroll
  for (int w = 1; w < 8; ++w) m = fmaxf(m, redA[w]);

  // ---- Pass 2: Z = sum exp(x-m), S1 = sum (x-m)*exp(x-m) ----
  float Z = 0.0f, S1 = 0.0f;
  for (int i = tid; i < NV4; i += TPB) {
    float4 v = b4[i];
    float x0 = v.x - m, x1 = v.y - m, x2 = v.z - m, x3 = v.w - m;
    float e0 = __expf(x0), e1 = __expf(x1), e2 = __expf(x2), e3 = __expf(x3);
    Z  += (e0 + e1) + (e2 + e3);
    S1  = fmaf(x0, e0, S1);
    S1  = fmaf(x1, e1, S1);
    S1  = fmaf(x2, e2, S1);
    S1  = fmaf(x3, e3, S1);
  }
  Z  = wave_sum32(Z);
  S1 = wave_sum32(S1);
  if (lane == 0) { redB[wid] = Z; redC[wid] = S1; }
  __syncthreads();

  if (tid == 0) {
    float z = 0.0f, s1 = 0.0f;
#pragma unroll
    for (int w = 0; w < 8; ++w) { z += redB[w]; s1 += redC[w]; }

    const int idx  = b * SEQ + (s + 1);          // shifted labels / mask
    const int lbl  = labels[idx];
    const int msk  = amask[idx];
    const bool valid = (lbl != -100) && (msk != 0);
    const int safe = (lbl == -100) ? 0 : lbl;

    const float xl    = buf[safe];
    const float logZ  = __logf(z);
    float tl          = (m + logZ) - xl;         // -log_softmax at label
    if (lbl == -100) tl = 0.0f;

    const float ent = logZ - s1 / z;             // -sum p*logp
    const float ne  = ent * (1.0f / LOG_V);      // normalized entropy
    const float sr  = tl / (ne + EPS_W);         // shock ratio
    const float sf  = fminf(sr * (1.0f / SHOCK_THRESHOLD), 10.0f);
    const float w   = (sr > SHOCK_THRESHOLD) ? (1.0f + ALPHA * sf) : ne;

    ws_sum[row] = valid ? (tl * w) : 0.0f;
    ws_cnt[row] = valid ? 1.0f : 0.0f;
  }
}

__global__ __launch_bounds__(TPB)
void reaft_final_kernel(const float* __restrict__ ws_sum,
                        const float* __restrict__ ws_cnt,
                        float*       __restrict__ out)
{
  __shared__ float rs[8];
  __shared__ float rc[8];
  float s = 0.0f, c = 0.0f;
  for (int i = threadIdx.x; i < NROWS; i += TPB) { s += ws_sum[i]; c += ws_cnt[i]; }
  s = wave_sum32(s);
  c = wave_sum32(c);
  const int lane = threadIdx.x & 31;
  const int wid  = threadIdx.x >> 5;
  if (lane == 0) { rs[wid] = s; rc[wid] = c; }
  __syncthreads();
  if (threadIdx.x == 0) {
    float S = 0.0f, C = 0.0f;
#pragma unroll
    for (int w = 0; w < 8; ++w) { S += rs[w]; C += rc[w]; }
    out[0] = S / fmaxf(C, 1.0f);
  }
}

extern "C" void kernel_launch(void* const* d_in, const int* in_sizes, int n_in,
                              void* d_out, int out_size, void* d_ws, size_t ws_size,
                              hipStream_t stream)
{
  const float* logits = (const float*)d_in[0];
  const int*   labels = (const int*)d_in[1];   // JAX default (no x64) -> int32
  const int*   amask  = (const int*)d_in[2];

  float* ws_sum = (float*)d_ws;
  float* ws_cnt = ws_sum + NROWS;              // 2*4094 floats = 32752 B of d_ws

  reaft_row_kernel<<<NROWS, TPB, 0, stream>>>(logits, labels, amask, ws_sum, ws_cnt);
  reaft_final_kernel<<<1, TPB, 0, stream>>>(ws_sum, ws_cnt, (float*)d_out);
}